// RegionalCluesExtractor_15925738733862
// MI455X (gfx1250) — compile-verified
//
#include <hip/hip_runtime.h>
#include <hip/hip_bf16.h>

typedef __attribute__((ext_vector_type(2))) float v2f;
typedef __attribute__((ext_vector_type(8))) float v8f;

#define B_   8
#define H_   128
#define W_   128
#define D_   512
#define NPIX (H_ * W_)   // 16384
#define NREG 16

// -------- Pass 0: normalized Gaussian weight table --------------------------
// Wn[pix][r] = 0.6 * Wmat[r,pix] / (wsum[pix] + 1e-8), pixel-major so the
// WMMA A-tile load is a contiguous v2f per lane.
__global__ void k_weights(float* __restrict__ Wn) {
    int pix = blockIdx.x * blockDim.x + threadIdx.x;
    if (pix >= NPIX) return;
    int h = pix >> 7, w = pix & 127;
    float hc = 3.0f * (float)h / 127.0f;   // linspace(0,3,128)
    float wc = 3.0f * (float)w / 127.0f;
    float hw[4], ww[4];
    float hs = 0.f, wsm = 0.f;
#pragma unroll
    for (int i = 0; i < 4; ++i) {
        float dh = (hc - (float)i) * 0.25f;           // (coord - i)/N
        hw[i] = __expf(-2.0f * dh * dh);
        hs += hw[i];
        float dw = (wc - (float)i) * 0.25f;
        ww[i] = __expf(-2.0f * dw * dw);
        wsm += ww[i];
    }
    float scale = 0.6f / (hs * wsm + 1e-8f);          // fold REGIONAL_WEIGHT + wsum
#pragma unroll
    for (int i = 0; i < 4; ++i)
#pragma unroll
        for (int j = 0; j < 4; ++j)
            Wn[(size_t)pix * 16 + i * 4 + j] = hw[i] * ww[j] * scale;
}

// -------- Pass A: 8x8 spatial block sums (one full read of features) --------
// bsum[b][bh][bw][d], 8*16*16*512 floats = 4 MB.
__global__ void __launch_bounds__(256) k_blocksum(const float* __restrict__ F,
                                                  float* __restrict__ bsum) {
    int bid = blockIdx.x;                 // b*256 + bh*16 + bw
    int b = bid >> 8, bh = (bid >> 4) & 15, bw = bid & 15;
    int tid = threadIdx.x;
    size_t base = (((size_t)b * H_ + bh * 8) * W_ + bw * 8) * D_;
    float s0 = 0.f, s1 = 0.f;
#pragma unroll 4
    for (int p = 0; p < 64; ++p) {
        int ph = p >> 3, pw = p & 7;
        size_t a = base + (size_t)(ph * W_ + pw) * D_;
        s0 += F[a + tid];
        s1 += F[a + tid + 256];
    }
    bsum[(size_t)bid * D_ + tid]       = s0;
    bsum[(size_t)bid * D_ + tid + 256] = s1;
}

// -------- Pass B: region means (x 1/1024) and 0.4 * global mean -------------
// Region i starts at row 24*i, spans 32 rows -> block rows [3i, 3i+4), same for cols.
__global__ void k_means(const float* __restrict__ bsum,
                        float* __restrict__ means, float* __restrict__ gmean) {
    int b = blockIdx.x >> 1;
    int d = ((blockIdx.x & 1) << 8) + threadIdx.x;
    const float* bs = bsum + (size_t)b * 256 * D_ + d;
    float g = 0.f;
    for (int blk = 0; blk < 256; ++blk) g += bs[(size_t)blk * D_];
    gmean[(size_t)b * D_ + d] = g * (0.4f / 16384.0f);   // (1-rw) * mean folded
#pragma unroll
    for (int i = 0; i < 4; ++i)
#pragma unroll
        for (int j = 0; j < 4; ++j) {
            float s = 0.f;
#pragma unroll
            for (int ph = 0; ph < 4; ++ph)
#pragma unroll
                for (int pw = 0; pw < 4; ++pw)
                    s += bs[(size_t)((3 * i + ph) * 16 + (3 * j + pw)) * D_];
            means[((size_t)b * NREG + i * 4 + j) * D_ + d] = s * (1.0f / 1024.0f);
        }
}

// -------- Pass C: WMMA broadcast-aggregate + fused epilogue -----------------
// One wave per 16-pixel x 16-channel tile.
// agg_tile(16x16) = Wn_tile(16 pix x 16 reg) @ means(16 reg x 16 d)
// via 4 chained V_WMMA_F32_16X16X4_F32 (K = 16 in chunks of 4).
__global__ void __launch_bounds__(256) k_main(const float* __restrict__ F,
                                              const float* __restrict__ Wn,
                                              const float* __restrict__ means,
                                              const float* __restrict__ gmean,
                                              float* __restrict__ out) {
    int lane = threadIdx.x & 31;
    int tile = blockIdx.x * 8 + (threadIdx.x >> 5);
    if (tile >= B_ * (NPIX / 16) * (D_ / 16)) return;   // uniform per wave

    int b     = tile >> 15;          // 1024 pixel-tiles * 32 d-tiles per batch
    int t     = tile & 32767;
    int pix0  = (t >> 5) * 16;       // d-tile fastest: waves in a block share A
    int d0    = (t & 31) * 16;

    int row = lane & 15;             // M (A) / N (B,C,D) position
    int hi  = lane >> 4;             // low/high half-wave

    // A: 16x4 f32 layout -> lane<16 holds K0(v0),K1(v1); lane>=16 K2,K3.
    const float* wn_row = Wn + (size_t)(pix0 + row) * 16 + hi * 2;
    // B: 4x16 f32 layout -> v0: K=(0|2) N=row ; v1: K=(1|3) N=row.
    const float* mb = means + (size_t)b * NREG * D_ + d0 + row;

    v8f acc = {0.f, 0.f, 0.f, 0.f, 0.f, 0.f, 0.f, 0.f};
#pragma unroll
    for (int kk = 0; kk < 4; ++kk) {
        v2f a = *(const v2f*)(wn_row + kk * 4);        // contiguous, 8B aligned
        int k0 = kk * 4 + hi * 2;
        v2f bm;
        bm.x = mb[(size_t)k0 * D_];
        bm.y = mb[(size_t)(k0 + 1) * D_];
        acc = __builtin_amdgcn_wmma_f32_16x16x4_f32(
            /*neg_a=*/false, a, /*neg_b=*/false, bm,
            /*c_mod=*/(short)0, acc, /*reuse_a=*/false, /*reuse_b=*/false);
    }

    float g = gmean[(size_t)b * D_ + d0 + row];        // already 0.4 * mean
    // C/D layout: VGPR v -> pixel row M = v + 8*hi, channel N = row.
    size_t fbase = ((size_t)b * NPIX + pix0 + hi * 8) * D_ + d0 + row;
#pragma unroll
    for (int v = 0; v < 8; ++v) {
        size_t a = fbase + (size_t)v * D_;
        out[a] = F[a] + acc[v] + g;
    }
}

extern "C" void kernel_launch(void* const* d_in, const int* in_sizes, int n_in,
                              void* d_out, int out_size, void* d_ws, size_t ws_size,
                              hipStream_t stream) {
    const float* F = (const float*)d_in[0];
    float* out = (float*)d_out;
    float* ws  = (float*)d_ws;
    // workspace layout (floats): ~5.3 MB total
    float* Wn    = ws;                          // 16384*16      = 262144
    float* bsum  = Wn + 262144;                 // 8*256*512     = 1048576
    float* means = bsum + 1048576;              // 8*16*512      = 65536
    float* gmean = means + 65536;               // 8*512         = 4096

    k_weights <<<NPIX / 256, 256, 0, stream>>>(Wn);
    k_blocksum<<<B_ * 256,   256, 0, stream>>>(F, bsum);
    k_means   <<<B_ * 2,     256, 0, stream>>>(bsum, means, gmean);
    // 8 batches * 1024 pixel-tiles * 32 d-tiles = 262144 waves, 8 waves/block
    k_main    <<<32768,      256, 0, stream>>>(F, Wn, means, gmean, out);
}